// OneClassLoss_8675833938190
// MI455X (gfx1250) — compile-verified
//
#include <hip/hip_runtime.h>
#include <math.h>

// ---------------------------------------------------------------------------
// OneClassLoss for MI455X (gfx1250, wave32).
//  * Dist GEMM (128x16384x128) via V_WMMA_F32_16X16X4_F32, split-K (2048 waves)
//  * 2D FFT as DFT-matrix matmuls (F symmetric), fp32 WMMA, per-image waves
//  * X panels staged to LDS with gfx1250 async-to-LDS ops
//  * wave-uniform tile indices forced to SGPRs (scalar loop control, EXEC=all-1)
//  * tile outputs stored transposed -> 2x global_store_b128 per lane
// ---------------------------------------------------------------------------

typedef __attribute__((ext_vector_type(2))) float v2f;
typedef __attribute__((ext_vector_type(4))) float v4f;
typedef __attribute__((ext_vector_type(8))) float v8f;

#define HW     128
#define BS     128
#define D      16384      // HW*HW
#define CHUNK  16         // images per FFT chunk
#define KSPLIT 32         // split-K factor for the Dist GEMM

// fp32 WMMA A/B fragment loader. A-matrix 16x4 layout (ISA 7.12.2):
//   lanes 0-15:  M = lane,    VGPR0/1 = K0,K1
//   lanes 16-31: M = lane-16, VGPR0/1 = K2,K3
// B (4x16) mirrored with N = lane. One contiguous float2 per lane when the
// "row" matrix is row-major.
__device__ __forceinline__ v2f load_frag(const float* __restrict__ base,
                                         int rowStart, int ld, int k, int lane) {
    int r  = lane & 15;
    int ko = (lane >> 4) << 1;
    const float* p = base + (size_t)(rowStart + r) * ld + k + ko;
    return v2f{p[0], p[1]};
}

__device__ __forceinline__ v8f wmma_f32(v2f a, v2f b, v8f c) {
    return __builtin_amdgcn_wmma_f32_16x16x4_f32(
        /*neg_a=*/false, a, /*neg_b=*/false, b,
        /*c_mod=*/(short)0, c, /*reuse_a=*/false, /*reuse_b=*/false);
}

// --------------------------- init: DFT matrix -------------------------------
__global__ __launch_bounds__(256) void init_w(float* __restrict__ Wr,
                                              float* __restrict__ Wi) {
    int idx = blockIdx.x * 256 + threadIdx.x;      // 0..16383
    int j = idx >> 7, k = idx & 127;
    int t = (j * k) & 127;                         // exact phase mod N
    float ang = -6.283185307179586f * (float)t * (1.0f / 128.0f);
    Wr[idx] = cosf(ang);
    Wi[idx] = sinf(ang);
}

// --------------------------- row squared norms ------------------------------
__global__ __launch_bounds__(256) void row_norms(const float* __restrict__ x1,
                                                 const float* __restrict__ x2,
                                                 float* __restrict__ na,
                                                 float* __restrict__ nb) {
    int row = blockIdx.x;                          // 0..255 (a rows then b rows)
    const float* p = (row < BS) ? (x1 + (size_t)row * D)
                                : (x2 + (size_t)(row - BS) * D);
    float s = 0.0f;
    for (int i = threadIdx.x; i < D; i += 256) { float v = p[i]; s += v * v; }
    __shared__ float red[256];
    red[threadIdx.x] = s; __syncthreads();
    for (int off = 128; off; off >>= 1) {
        if (threadIdx.x < (unsigned)off) red[threadIdx.x] += red[threadIdx.x + off];
        __syncthreads();
    }
    if (threadIdx.x == 0) {
        if (row < BS) na[row] = red[0]; else nb[row - BS] = red[0];
    }
}

// --------------------------- split-K GEMM: Gpart[s] = (A.B^T)^T slice -------
// 64 tiles x KSPLIT splits = 2048 waves (512 blocks x 4 waves).
__global__ __launch_bounds__(128) void dist_gemm(const float* __restrict__ A,
                                                 const float* __restrict__ B,
                                                 float* __restrict__ Gpart) {
    // wave-uniform ids into SGPRs -> scalar loop control, EXEC stays all-ones
    int gw   = __builtin_amdgcn_readfirstlane(
                   (blockIdx.x * blockDim.x + threadIdx.x) >> 5);  // 0..2047
    int lane = threadIdx.x & 31;
    int tile  = gw & 63;
    int split = gw >> 6;                                          // 0..KSPLIT-1
    int tm = (tile >> 3) << 4;
    int tn = (tile & 7) << 4;
    int k0 = split * (D / KSPLIT);
    int k1 = k0 + (D / KSPLIT);

    v8f acc = {};
    #pragma unroll 4
    for (int k = k0; k < k1; k += 4) {
        v2f af = load_frag(A, tm, D, k, lane);   // rows of a
        v2f bf = load_frag(B, tn, D, k, lane);   // rows of b (B^T columns)
        acc = wmma_f32(af, bf, acc);
    }

    // store transposed: Gt[n][m]; each lane owns 8 contiguous floats
    float* Gp = Gpart + (size_t)split * D;
    int n     = tn + (lane & 15);
    int mbase = tm + ((lane >> 4) << 3);
    float* dst = Gp + n * HW + mbase;
    *(v4f*)dst       = v4f{acc[0], acc[1], acc[2], acc[3]};
    *((v4f*)dst + 1) = v4f{acc[4], acc[5], acc[6], acc[7]};
}

// --------------------------- reduce split-K, form Dist ----------------------
// Gpart is stored transposed: linear idx corresponds to (n, m).
__global__ __launch_bounds__(256) void dist_finalize(const float* __restrict__ Gpart,
                                                     const float* __restrict__ na,
                                                     const float* __restrict__ nb,
                                                     float* __restrict__ Dist) {
    int idx = blockIdx.x * 256 + threadIdx.x;    // 0..16383, = n*HW + m
    float g = 0.0f;
    #pragma unroll 8
    for (int s = 0; s < KSPLIT; ++s) g += Gpart[(size_t)s * D + idx];
    int n = idx >> 7, m = idx & 127;
    float d = na[m] + nb[n] - 2.0f * g;
    Dist[m * HW + n] = sqrtf(fmaxf(d, 0.0f));
}

// --------------------------- CE on diagonal of log_softmax ------------------
__global__ __launch_bounds__(128) void ce_kernel(const float* __restrict__ Dist,
                                                 float* __restrict__ ceOut) {
    int i = threadIdx.x;                          // one row per thread
    const float* row = Dist + i * HW;
    float mx = -INFINITY;
    for (int j = 0; j < HW; ++j) mx = fmaxf(mx, row[j]);
    float s = 0.0f;
    for (int j = 0; j < HW; ++j) s += expf(row[j] - mx);
    float lse = mx + logf(s);
    __shared__ float red[128];
    red[i] = lse - row[i];                        // -log_softmax diag term
    __syncthreads();
    for (int off = 64; off; off >>= 1) {
        if (i < off) red[i] += red[i + off];
        __syncthreads();
    }
    if (i == 0) ceOut[0] = red[0] * (1.0f / BS);
}

// --------------------------- FFT stage 1: T = X * F (store T^T) -------------
// Block = 4 waves sharing one 16-row X panel, staged into LDS via the gfx1250
// async-to-LDS path. Grid: CHUNK * 8 tm * 2 halves = CHUNK*16 blocks.
__global__ __launch_bounds__(128) void fft_stage1(const float* __restrict__ x1,
                                                  const float* __restrict__ x2,
                                                  const float* __restrict__ Wr,
                                                  const float* __restrict__ Wi,
                                                  float* __restrict__ T,
                                                  int imgBase) {
    int b     = blockIdx.x;
    int img   = b >> 4;                           // 0..CHUNK-1
    int rem   = b & 15;
    int tmIdx = rem >> 1;                         // 0..7
    int half  = rem & 1;                          // 0..1
    int wv    = __builtin_amdgcn_readfirstlane(threadIdx.x >> 5);  // 0..3, SGPR
    int lane  = threadIdx.x & 31;
    int tm = tmIdx << 4;
    int tn = ((half << 2) + wv) << 4;

    int gimg = imgBase + img;
    const float* X = (gimg < BS) ? (x1 + (size_t)gimg * D)
                                 : (x2 + (size_t)(gimg - BS) * D);

    // Stage the contiguous 8 KB X panel (rows tm..tm+15) into LDS with
    // GLOBAL_LOAD_ASYNC_TO_LDS_B128 (VDST = LDS byte addr, VADDR = 64b global).
    __shared__ float xs[16 * HW];
    {
        unsigned           ldsBase = (unsigned)(size_t)(void*)xs;
        unsigned long long gBase   = (unsigned long long)(size_t)(X + (size_t)tm * HW);
        #pragma unroll
        for (int i = 0; i < 4; ++i) {
            unsigned           off   = (unsigned)((threadIdx.x + i * 128u) * 16u);
            unsigned           laddr = ldsBase + off;
            unsigned long long gaddr = gBase + off;
            asm volatile("global_load_async_to_lds_b128 %0, %1, off"
                         :: "v"(laddr), "v"(gaddr) : "memory");
        }
        asm volatile("s_wait_asynccnt 0" ::: "memory");
        __syncthreads();
    }

    v8f ar = {}, ai = {};
    #pragma unroll 4
    for (int k = 0; k < HW; k += 4) {
        int r = lane & 15, ko = (lane >> 4) << 1;
        const float* px = xs + r * HW + k + ko;   // A-frag from LDS (ds_load_b64)
        v2f xf = v2f{px[0], px[1]};
        v2f wr = load_frag(Wr, tn, HW, k, lane);  // W symmetric: B(k,n)=W[n][k]
        v2f wi = load_frag(Wi, tn, HW, k, lane);
        ar = wmma_f32(xf, wr, ar);
        ai = wmma_f32(xf, wi, ai);
    }

    // store transposed: Tt[n][m] = T[m][n]; each lane owns 8 contiguous floats
    float* Tr = T + (size_t)img * 2 * D;
    float* Ti = Tr + D;
    int n     = tn + (lane & 15);
    int mbase = tm + ((lane >> 4) << 3);
    float* dr = Tr + n * HW + mbase;
    float* di = Ti + n * HW + mbase;
    *(v4f*)dr       = v4f{ar[0], ar[1], ar[2], ar[3]};
    *((v4f*)dr + 1) = v4f{ar[4], ar[5], ar[6], ar[7]};
    *(v4f*)di       = v4f{ai[0], ai[1], ai[2], ai[3]};
    *((v4f*)di + 1) = v4f{ai[4], ai[5], ai[6], ai[7]};
}

// --------------------------- FFT stage 2: |F*T|^2 into per-image psd --------
// One wave per (image, tile): 1024 waves per chunk launch, no atomics.
// psdPart kept transposed (psd[n][m]) -> contiguous b128 RMW per lane; the
// final log/mean reduction is permutation-invariant so no unpermute needed.
__global__ __launch_bounds__(128) void fft_stage2(const float* __restrict__ Wr,
                                                  const float* __restrict__ Wi,
                                                  const float* __restrict__ T,
                                                  float* __restrict__ psdPart,
                                                  int first) {
    int gw   = __builtin_amdgcn_readfirstlane(
                   (blockIdx.x * blockDim.x + threadIdx.x) >> 5);  // 0..1023
    int lane = threadIdx.x & 31;
    int img  = gw >> 6;                           // 0..CHUNK-1 (local)
    int tile = gw & 63;
    int tm = (tile >> 3) << 4;
    int tn = (tile & 7) << 4;

    const float* Tr = T + (size_t)img * 2 * D;
    const float* Ti = Tr + D;
    v8f p = {}, q = {}, u = {}, s = {};
    #pragma unroll 2
    for (int k = 0; k < HW; k += 4) {
        v2f awr = load_frag(Wr, tm, HW, k, lane); // A rows of W
        v2f awi = load_frag(Wi, tm, HW, k, lane);
        v2f btr = load_frag(Tr, tn, HW, k, lane); // Tt row-major
        v2f bti = load_frag(Ti, tn, HW, k, lane);
        p = wmma_f32(awr, btr, p);                // Wr*Tr
        q = wmma_f32(awi, bti, q);                // Wi*Ti
        u = wmma_f32(awr, bti, u);                // Wr*Ti
        s = wmma_f32(awi, btr, s);                // Wi*Tr
    }

    v8f vz;
    #pragma unroll
    for (int e = 0; e < 8; ++e) {
        float zr = p[e] - q[e];
        float zi = u[e] + s[e];
        vz[e] = zr * zr + zi * zi;
    }

    float* psd = psdPart + (size_t)img * D;       // per-image-slot partial
    int n     = tn + (lane & 15);
    int mbase = tm + ((lane >> 4) << 3);
    float* dst = psd + n * HW + mbase;            // transposed, contiguous
    v4f lo = v4f{vz[0], vz[1], vz[2], vz[3]};
    v4f hi = v4f{vz[4], vz[5], vz[6], vz[7]};
    if (first) {
        *(v4f*)dst       = lo;
        *((v4f*)dst + 1) = hi;
    } else {
        v4f olo = *(v4f*)dst;
        v4f ohi = *((v4f*)dst + 1);
        *(v4f*)dst       = olo + lo;              // slot owned per launch
        *((v4f*)dst + 1) = ohi + hi;
    }
}

// --------------------------- final: r = mean(log)-log(mean); out ------------
__global__ __launch_bounds__(256) void final_kernel(const float* __restrict__ psdPart,
                                                    const float* __restrict__ ceS,
                                                    float* __restrict__ out) {
    float sl = 0.0f, sv = 0.0f;
    for (int e = threadIdx.x; e < D; e += 256) {
        float acc = 0.0f;
        #pragma unroll
        for (int g = 0; g < CHUNK; ++g) acc += psdPart[(size_t)g * D + e];
        float v = acc * (1.0f / 256.0f);          // mean over 2*BS images
        sl += logf(v);
        sv += v;
    }
    __shared__ float rl[256], rv[256];
    rl[threadIdx.x] = sl; rv[threadIdx.x] = sv; __syncthreads();
    for (int off = 128; off; off >>= 1) {
        if (threadIdx.x < (unsigned)off) {
            rl[threadIdx.x] += rl[threadIdx.x + off];
            rv[threadIdx.x] += rv[threadIdx.x + off];
        }
        __syncthreads();
    }
    if (threadIdx.x == 0) {
        float r = rl[0] * (1.0f / D) - logf(rv[0] * (1.0f / D));
        out[0] = ceS[0] - 0.1f * r;
    }
}

// ---------------------------------------------------------------------------
extern "C" void kernel_launch(void* const* d_in, const int* in_sizes, int n_in,
                              void* d_out, int out_size, void* d_ws, size_t ws_size,
                              hipStream_t stream) {
    (void)in_sizes; (void)n_in; (void)out_size; (void)ws_size;
    const float* x1 = (const float*)d_in[0];
    const float* x2 = (const float*)d_in[1];
    float* out = (float*)d_out;

    // workspace layout (floats), ~5.2 MB total
    float* ws      = (float*)d_ws;
    float* Dist    = ws;                    // 16384
    float* na      = ws + 16384;            // 128
    float* nb      = ws + 16512;            // 128
    float* ceS     = ws + 16640;            // 1 (padded to 16896)
    float* Wr      = ws + 16896;            // 16384
    float* Wi      = Wr + D;                // 16384
    float* psdPart = Wi + D;                // CHUNK  * 16384
    float* Gpart   = psdPart + CHUNK * D;   // KSPLIT * 16384
    float* T       = Gpart + KSPLIT * D;    // CHUNK * 2 * 16384

    init_w       <<<64, 256, 0, stream>>>(Wr, Wi);
    row_norms    <<<256, 256, 0, stream>>>(x1, x2, na, nb);
    dist_gemm    <<<(64 * KSPLIT) / 4, 128, 0, stream>>>(x1, x2, Gpart);
    dist_finalize<<<64, 256, 0, stream>>>(Gpart, na, nb, Dist);
    ce_kernel    <<<1, 128, 0, stream>>>(Dist, ceS);

    for (int base = 0; base < 2 * BS; base += CHUNK) {
        fft_stage1<<<CHUNK * 16, 128, 0, stream>>>(x1, x2, Wr, Wi, T, base);
        fft_stage2<<<(CHUNK * 64) / 4, 128, 0, stream>>>(Wr, Wi, T, psdPart,
                                                         base == 0 ? 1 : 0);
    }

    final_kernel<<<1, 256, 0, stream>>>(psdPart, ceS, out);
}